// DisMult_13013750907174
// MI455X (gfx1250) — compile-verified
//
#include <hip/hip_runtime.h>
#include <hip/hip_bf16.h>

typedef __attribute__((ext_vector_type(2))) float v2f;
typedef __attribute__((ext_vector_type(8))) float v8f;

#define DIM 128
#define WAVES_PER_BLOCK 5   // 15625 tiles/relation = 5^6, divides exactly

// One wave32 scores 16 edges of one relation using fp32 WMMA (16x16x4).
// A[m,k] = u[edge_m] * rel, B[k,n] = v[edge_n]; diag(A x B) = scores.
__global__ __launch_bounds__(WAVES_PER_BLOCK * 32)
void distmult_wmma_f32_kernel(const float* __restrict__ node,
                              const float* __restrict__ rel,
                              const int*   __restrict__ src,
                              const int*   __restrict__ dst,
                              float*       __restrict__ out,
                              int E, int tilesPerRel) {
    __shared__ float srel[DIM];

    const int r = blockIdx.y;

    // Stage this relation's 128-float vector into LDS once per block.
    if (threadIdx.x < DIM)
        srel[threadIdx.x] = rel[r * DIM + threadIdx.x];
    __syncthreads();

    const int wave = threadIdx.x >> 5;
    const int lane = threadIdx.x & 31;
    const int tile = blockIdx.x * WAVES_PER_BLOCK + wave;
    if (tile >= tilesPerRel) return;   // uniform per-wave: EXEC stays all-ones inside

    const int e0  = tile << 4;         // 16 edges per tile
    const int l16 = lane & 15;         // edge slot this lane serves (both halves)
    const int hiOff = (lane >> 4) << 1; // low lanes: K+0..1, high lanes: K+2..3

    // Per-lane row pointers for the gathers (random, but table fits in 192MB L2).
    const long eidx = (long)r * E + e0 + l16;
    const float* uRow = node + (size_t)src[eidx] * DIM + hiOff;
    const float* vRow = node + (size_t)dst[eidx] * DIM + hiOff;
    const float* rRow = srel + hiOff;

    v8f c = {};   // 16x16 fp32 accumulator (diagonal = 16 scores)

#pragma unroll
    for (int k = 0; k < DIM; k += 4) {
        v2f u2 = *(const v2f*)(uRow + k);   // global_load_b64 (L2-resident gather)
        v2f w2 = *(const v2f*)(vRow + k);   // global_load_b64
        v2f r2 = *(const v2f*)(rRow + k);   // ds_load_b64
        v2f a2 = u2 * r2;                   // fold DistMult relation into A
        // D = A(16x4 f32) x B(4x16 f32) + C ; fp32 WMMA, RTNE accumulate
        c = __builtin_amdgcn_wmma_f32_16x16x4_f32(
                /*neg_a=*/false, a2,
                /*neg_b=*/false, w2,
                /*c_mod=*/(short)0, c,
                /*reuse_a=*/false, /*reuse_b=*/false);
    }

    // Diagonal extraction:
    //   m = 0..7  -> lane m,      component m
    //   m = 8..15 -> lane m+16,   component m-8
    int m = -1;
    if (lane < 8)        m = lane;
    else if (lane >= 24) m = lane - 16;

    if (m >= 0) {
        const int ci = lane & 7;   // component index in both cases
        float v = c[0];
        if (ci == 1) v = c[1];
        if (ci == 2) v = c[2];
        if (ci == 3) v = c[3];
        if (ci == 4) v = c[4];
        if (ci == 5) v = c[5];
        if (ci == 6) v = c[6];
        if (ci == 7) v = c[7];
        out[(size_t)r * E + (size_t)(e0 + m)] = v;
    }
}

extern "C" void kernel_launch(void* const* d_in, const int* in_sizes, int n_in,
                              void* d_out, int out_size, void* d_ws, size_t ws_size,
                              hipStream_t stream) {
    const float* node = (const float*)d_in[0];   // [N, 128] fp32
    const float* rel  = (const float*)d_in[1];   // [R, 128] fp32
    const int*   src  = (const int*)d_in[2];     // [R, E] int32
    const int*   dst  = (const int*)d_in[3];     // [R, E] int32
    float*       out  = (float*)d_out;           // [R, E] fp32

    const int R = in_sizes[1] / DIM;             // 4
    const int E = in_sizes[2] / R;               // 250000
    const int tilesPerRel = E / 16;              // 15625

    dim3 grid((tilesPerRel + WAVES_PER_BLOCK - 1) / WAVES_PER_BLOCK, R);
    dim3 block(WAVES_PER_BLOCK * 32);

    distmult_wmma_f32_kernel<<<grid, block, 0, stream>>>(
        node, rel, src, dst, out, E, tilesPerRel);
}